// GravNetBlock_14559939133841
// MI455X (gfx1250) — compile-verified
//
#include <hip/hip_runtime.h>
#include <hip/hip_bf16.h>
#include <math.h>

// ---- problem constants (match reference) ----
#define N_NODES 65536   // E*H
#define E_EV    32
#define H_HITS  2048
#define K_NN    40
#define C_IN    64
#define C_OUT   96
#define S_DIM   4
#define P_DIM   22
#define BN_EPS  1e-5f

typedef float v2f __attribute__((ext_vector_type(2)));
typedef float v4f __attribute__((ext_vector_type(4)));
typedef float v8f __attribute__((ext_vector_type(8)));

// CDNA5 fp32 WMMA: D(16x16) = A(16x4) * B(4x16) + C
__device__ __forceinline__ v8f wmma4(v2f a, v2f b, v8f c) {
  return __builtin_amdgcn_wmma_f32_16x16x4_f32(false, a, false, b, (short)0, c,
                                               false, false);
}

// ---------------- embeddings: s = x@Ws+bs, h = x@Wh+bh ----------------
__global__ void k_embed(const float* __restrict__ x,
                        const float* __restrict__ Ws, const float* __restrict__ bs,
                        const float* __restrict__ Wh, const float* __restrict__ bh,
                        float* __restrict__ s, float* __restrict__ h) {
  int t = blockIdx.x * blockDim.x + threadIdx.x;
  if (t >= N_NODES * (S_DIM + P_DIM)) return;
  int row = t / (S_DIM + P_DIM);
  int c   = t % (S_DIM + P_DIM);
  const float* xr = x + (size_t)row * C_IN;
  if (c < S_DIM) {
    float acc = bs[c];
    for (int j = 0; j < C_IN; ++j) acc += xr[j] * Ws[j * S_DIM + c];
    s[(size_t)row * S_DIM + c] = acc;
  } else {
    int p = c - S_DIM;
    float acc = bh[p];
    for (int j = 0; j < C_IN; ++j) acc += xr[j] * Wh[j * P_DIM + p];
    h[(size_t)row * P_DIM + p] = acc;
  }
}

// ---------------- kNN: WMMA distance tiles + top-40 selection ----------------
#define CHUNK 256
__global__ void __launch_bounds__(128)
k_knn(const float* __restrict__ s, int* __restrict__ idx, float* __restrict__ wv) {
  __shared__ float dchunk[16][CHUNK];
  __shared__ float td[16][K_NN];
  __shared__ int   ti[16][K_NN];
  __shared__ float snr[16];

  const int e = blockIdx.y;
  const int rowbase = blockIdx.x * 16;                  // hit index within event
  const float* se = s + (size_t)e * H_HITS * S_DIM;

  const int tid   = threadIdx.x;
  const int lane  = tid & 31;
  const int wave  = tid >> 5;
  const int m     = lane & 15;
  const bool lo   = (lane < 16);
  const int khalf = lo ? 0 : 2;

  for (int t = tid; t < 16 * K_NN; t += blockDim.x) {
    (&td[0][0])[t] = 1e30f;
    (&ti[0][0])[t] = 0;
  }
  if (tid < 16) {
    const float* sr = se + (size_t)(rowbase + tid) * S_DIM;
    snr[tid] = sr[0]*sr[0] + sr[1]*sr[1] + sr[2]*sr[2] + sr[3]*sr[3];
  }
  __syncthreads();

  // A tile (16x4, rows = queries): constant across the whole sweep
  v2f a = *(const v2f*)(se + (size_t)(rowbase + m) * S_DIM + khalf);
  // per-lane row norms for epilogue (invariant across all tiles) — hoisted
  const int hi = lo ? 0 : 8;
  float sr8[8];
#pragma unroll
  for (int v = 0; v < 8; ++v) sr8[v] = snr[v + hi];

  for (int cb = 0; cb < H_HITS; cb += CHUNK) {
    // 16 column tiles per chunk, 4 per wave; software-pipelined point loads
    v4f sc = *(const v4f*)(se + (size_t)(cb + wave * 64 + m) * S_DIM);
#pragma unroll
    for (int it = 0; it < 4; ++it) {
      const int tile = wave * 4 + it;
      v4f scn;
      if (it < 3)
        scn = *(const v4f*)(se + (size_t)(cb + (tile + 1) * 16 + m) * S_DIM);
      v2f b;
      b.x = lo ? sc.x : sc.z;                          // B = s_cols^T, lane select
      b.y = lo ? sc.y : sc.w;
      v8f acc = {};
      acc = wmma4(a, b, acc);                          // 16x16 dot-product tile
      const float snc = sc.x*sc.x + sc.y*sc.y + sc.z*sc.z + sc.w*sc.w;
#pragma unroll
      for (int v = 0; v < 8; ++v)
        dchunk[v + hi][tile * 16 + m] = sr8[v] + snc - 2.0f * acc[v];
      sc = scn;
    }
    __syncthreads();
    // insertion-sorted top-K per query row (one thread per row)
    if (tid < 16) {
      const int r = tid;
      float worst = td[r][K_NN - 1];
      for (int j = 0; j < CHUNK; ++j) {
        float d = dchunk[r][j];
        if (d < worst) {
          int p = K_NN - 1;
          while (p > 0 && td[r][p - 1] > d) {
            td[r][p] = td[r][p - 1];
            ti[r][p] = ti[r][p - 1];
            --p;
          }
          td[r][p] = d;
          ti[r][p] = cb + j;
          worst = td[r][K_NN - 1];
        }
      }
    }
    __syncthreads();
  }
  for (int t = tid; t < 16 * K_NN; t += blockDim.x) {
    const int r = t / K_NN, kk = t % K_NN;
    const size_t node = (size_t)e * H_HITS + rowbase + r;
    idx[node * K_NN + kk] = e * H_HITS + ti[r][kk];     // absolute node id
    wv[node * K_NN + kk]  = __expf(-10.0f * td[r][kk]);
  }
}

// ---------------- message aggregation: mean & max over K ----------------
__global__ void k_agg(const float* __restrict__ h, const int* __restrict__ idx,
                      const float* __restrict__ wv, float* __restrict__ agg) {
  int t = blockIdx.x * blockDim.x + threadIdx.x;
  if (t >= N_NODES * P_DIM) return;
  const int n = t / P_DIM, p = t % P_DIM;
  const int*   id = idx + (size_t)n * K_NN;
  const float* w  = wv  + (size_t)n * K_NN;
  float sum = 0.f, mx = -1e30f;
  for (int k = 0; k < K_NN; ++k) {
    const float msg = h[(size_t)id[k] * P_DIM + p] * w[k];
    sum += msg;
    mx = fmaxf(mx, msg);
  }
  agg[(size_t)n * (2 * P_DIM) + p]         = sum * (1.0f / K_NN);
  agg[(size_t)n * (2 * P_DIM) + P_DIM + p] = mx;
}

// ---------------- generic fused WMMA GEMM ----------------
// out[M,N] = act( A[M,K]@W[K,N] (+ A2@W2) (+ bias) (+ rowbias[row>>11]) )
// Software-pipelined K loop: next step's A/B loads issue before current WMMA.
template <bool HAS2, bool TANH, bool ROWBIAS>
__global__ void __launch_bounds__(128)
k_gemm(const float* __restrict__ A, const float* __restrict__ W, int K,
       const float* __restrict__ A2, const float* __restrict__ W2, int K2,
       const float* __restrict__ bias, const float* __restrict__ rowbias,
       float* __restrict__ out, int N) {
  const int lane = threadIdx.x & 31;
  const int wave = threadIdx.x >> 5;
  const int colTile = blockIdx.y * 4 + wave;
  if (colTile * 16 >= N) return;                 // wave-uniform (EXEC stays all-1s)
  const int m     = lane & 15;
  const int khalf = (lane >> 4) << 1;
  const int row   = blockIdx.x * 16 + m;
  const int col   = colTile * 16 + m;

  v8f c = {};
  {
    const float* Ar = A + (size_t)row * K;
    __builtin_prefetch(Ar, 0, 3);                // global_prefetch_b8
    v2f a = *(const v2f*)(Ar + khalf);
    v2f b;
    b.x = W[(size_t)khalf * N + col];
    b.y = W[(size_t)(khalf + 1) * N + col];
    for (int k = 4; k < K; k += 4) {
      v2f an = *(const v2f*)(Ar + k + khalf);
      v2f bn;
      bn.x = W[(size_t)(k + khalf)     * N + col];
      bn.y = W[(size_t)(k + khalf + 1) * N + col];
      c = wmma4(a, b, c);
      a = an; b = bn;
    }
    c = wmma4(a, b, c);
  }
  if (HAS2) {
    const float* Ar = A2 + (size_t)row * K2;
    __builtin_prefetch(Ar, 0, 3);
    v2f a = *(const v2f*)(Ar + khalf);
    v2f b;
    b.x = W2[(size_t)khalf * N + col];
    b.y = W2[(size_t)(khalf + 1) * N + col];
    for (int k = 4; k < K2; k += 4) {
      v2f an = *(const v2f*)(Ar + k + khalf);
      v2f bn;
      bn.x = W2[(size_t)(k + khalf)     * N + col];
      bn.y = W2[(size_t)(k + khalf + 1) * N + col];
      c = wmma4(a, b, c);
      a = an; b = bn;
    }
    c = wmma4(a, b, c);
  }
  const int hi = (lane >> 4) << 3;
#pragma unroll
  for (int v = 0; v < 8; ++v) {
    const int r = blockIdx.x * 16 + v + hi;
    float val = c[v];
    if (bias) val += bias[col];
    if (ROWBIAS) val += rowbias[(size_t)(r >> 11) * N + col];  // event = row/2048
    if (TANH) val = tanhf(val);
    out[(size_t)r * N + col] = val;
  }
}

// ---------------- BatchNorm: stats + apply ----------------
__global__ void k_bn_stats(const float* __restrict__ X, int C, int rowsPerBlock,
                           float* __restrict__ stats /* [2C]: sum, sumsq */) {
  const int c  = threadIdx.x;                      // blockDim == C
  const int r0 = blockIdx.x * rowsPerBlock;
  float s = 0.f, q = 0.f;
  for (int i = 0; i < rowsPerBlock; ++i) {
    const float v = X[(size_t)(r0 + i) * C + c];
    s += v; q += v * v;
  }
  atomicAdd(&stats[c], s);
  atomicAdd(&stats[C + c], q);
}

__global__ void k_bn_apply(const float* __restrict__ X, const float* __restrict__ stats,
                           const float* __restrict__ g, const float* __restrict__ b,
                           float* __restrict__ Y, int C, int total) {
  int t = blockIdx.x * blockDim.x + threadIdx.x;
  if (t >= total) return;
  const int c = t % C;
  const float m   = stats[c] * (1.0f / N_NODES);
  const float var = stats[C + c] * (1.0f / N_NODES) - m * m;
  Y[t] = (X[t] - m) * rsqrtf(var + BN_EPS) * g[c] + b[c];
}

// ---------------- global exchange helpers ----------------
__global__ void k_event_stats(const float* __restrict__ Y, float* __restrict__ estats) {
  const int e = blockIdx.x, c = threadIdx.x;       // blockDim == 96
  const float* base = Y + (size_t)e * H_HITS * C_OUT;
  float s = 0.f, mn = 1e30f, mx = -1e30f;
  for (int i = 0; i < H_HITS; ++i) {
    const float v = base[(size_t)i * C_OUT + c];
    s += v; mn = fminf(mn, v); mx = fmaxf(mx, v);
  }
  estats[e * 288 + c]       = s * (1.0f / H_HITS);  // mean
  estats[e * 288 + 96 + c]  = mn;                    // min
  estats[e * 288 + 192 + c] = mx;                    // max
}

// bvec[e][n] = b3[n] + stats[e][0:288] @ W3[0:288][n]
__global__ void k_bvec(const float* __restrict__ estats, const float* __restrict__ W3,
                       const float* __restrict__ b3, float* __restrict__ bvec) {
  const int e = blockIdx.x, n = threadIdx.x;        // blockDim == 96
  float acc = b3[n];
  const float* st = estats + e * 288;
  for (int j = 0; j < 288; ++j) acc += st[j] * W3[(size_t)j * 96 + n];
  bvec[e * 96 + n] = acc;
}

// ================= host-side launch =================
extern "C" void kernel_launch(void* const* d_in, const int* in_sizes, int n_in,
                              void* d_out, int out_size, void* d_ws, size_t ws_size,
                              hipStream_t stream) {
  // setup_inputs dict order: x, batch, then params in insertion order
  const float* x     = (const float*)d_in[0];   // [N,64]
  // d_in[1] = batch (int) — event id is row>>11, not needed
  const float* Ws    = (const float*)d_in[2];   // [64,4]
  const float* bs    = (const float*)d_in[3];
  const float* Wh    = (const float*)d_in[4];   // [64,22]
  const float* bh    = (const float*)d_in[5];
  const float* Wo1   = (const float*)d_in[6];   // [64,96]
  const float* Wo2   = (const float*)d_in[7];   // [44,96]
  const float* bo2   = (const float*)d_in[8];
  const float* bn1g  = (const float*)d_in[9];
  const float* bn1b  = (const float*)d_in[10];
  const float* W1    = (const float*)d_in[11];  // [96,128]
  const float* b1    = (const float*)d_in[12];
  const float* bn2g  = (const float*)d_in[13];
  const float* bn2b  = (const float*)d_in[14];
  const float* W2    = (const float*)d_in[15];  // [128,96]
  const float* b2    = (const float*)d_in[16];
  const float* W3    = (const float*)d_in[17];  // [384,96] rows: mean|min|max|x
  const float* b3    = (const float*)d_in[18];
  const float* bn3g  = (const float*)d_in[19];
  const float* bn3b  = (const float*)d_in[20];
  (void)in_sizes; (void)n_in; (void)out_size;

  const size_t N = N_NODES;
  float* s     = (float*)d_ws;                  // N*4
  float* h     = s + N * 4;                     // N*22
  int*   idx   = (int*)(h + N * 22);            // N*40
  float* wv    = (float*)(idx + N * 40);        // N*40
  float* agg   = wv + N * 40;                   // N*44
  float* bufA  = agg + N * 44;                  // N*128
  float* bufB  = bufA + N * 128;                // N*128
  float* stats = bufB + N * 128;                // 256
  float* est   = stats + 256;                   // 32*288
  float* bvec  = est + 32 * 288;                // 32*96
  (void)ws_size;

  // 1. embeddings
  {
    int tot = N_NODES * (S_DIM + P_DIM);
    k_embed<<<(tot + 255) / 256, 256, 0, stream>>>(x, Ws, bs, Wh, bh, s, h);
  }
  // 2. kNN (WMMA distances + top-40)
  k_knn<<<dim3(H_HITS / 16, E_EV), 128, 0, stream>>>(s, idx, wv);
  // 3. aggregation
  {
    int tot = N_NODES * P_DIM;
    k_agg<<<(tot + 255) / 256, 256, 0, stream>>>(h, idx, wv, agg);
  }
  const dim3 gemmBlk(128);
  const dim3 grid96(N_NODES / 16, 2);   // 6 col tiles (2 idle waves in 2nd slab)
  const dim3 grid128(N_NODES / 16, 2);  // 8 col tiles
  // 4. y0 = x@Wo1 + agg@Wo2 + bo2
  k_gemm<true, false, false><<<grid96, gemmBlk, 0, stream>>>(
      x, Wo1, C_IN, agg, Wo2, 2 * P_DIM, bo2, nullptr, bufA, 96);
  // 5. BN1 -> bufB
  hipMemsetAsync(stats, 0, 2 * 96 * sizeof(float), stream);
  k_bn_stats<<<N_NODES / 512, 96, 0, stream>>>(bufA, 96, 512, stats);
  k_bn_apply<<<(N_NODES * 96 + 255) / 256, 256, 0, stream>>>(
      bufA, stats, bn1g, bn1b, bufB, 96, N_NODES * 96);
  // 6. y2 = tanh(bufB@W1 + b1) -> bufA [N,128]
  k_gemm<false, true, false><<<grid128, gemmBlk, 0, stream>>>(
      bufB, W1, 96, nullptr, nullptr, 0, b1, nullptr, bufA, 128);
  // 7. BN2 -> bufB
  hipMemsetAsync(stats, 0, 2 * 128 * sizeof(float), stream);
  k_bn_stats<<<N_NODES / 512, 128, 0, stream>>>(bufA, 128, 512, stats);
  k_bn_apply<<<(N_NODES * 128 + 255) / 256, 256, 0, stream>>>(
      bufA, stats, bn2g, bn2b, bufB, 128, N_NODES * 128);
  // 8. y4 = tanh(bufB@W2 + b2) -> bufA [N,96]
  k_gemm<false, true, false><<<grid96, gemmBlk, 0, stream>>>(
      bufB, W2, 128, nullptr, nullptr, 0, b2, nullptr, bufA, 96);
  // 9. global exchange folded into per-event row bias
  k_event_stats<<<E_EV, 96, 0, stream>>>(bufA, est);
  k_bvec<<<E_EV, 96, 0, stream>>>(est, W3, b3, bvec);
  // 10. y5 = tanh(bufA@W3[288:384] + bvec[event]) -> bufB [N,96]
  k_gemm<false, true, true><<<grid96, gemmBlk, 0, stream>>>(
      bufA, W3 + 288 * 96, 96, nullptr, nullptr, 0, nullptr, bvec, bufB, 96);
  // 11. BN3 -> d_out
  hipMemsetAsync(stats, 0, 2 * 96 * sizeof(float), stream);
  k_bn_stats<<<N_NODES / 512, 96, 0, stream>>>(bufB, 96, 512, stats);
  k_bn_apply<<<(N_NODES * 96 + 255) / 256, 256, 0, stream>>>(
      bufB, stats, bn3g, bn3b, (float*)d_out, 96, N_NODES * 96);
}